// EdgeGCN_54846732369958
// MI455X (gfx1250) — compile-verified
//
#include <hip/hip_runtime.h>

// ---------------------------------------------------------------------------
// EdgeGCN: out[i*N+j,:] = (feat[j]-feat[i])@W0 + adj[i,j]*Wlast + bias
// Factored: P = feat @ W0 (WMMA fp32), then stream P[j]-P[i]+adj*Wlast+bias.
// HBM traffic ~1.09 GB (1 GiB NT stores + 16 MB adj); P (512KB) stays in L2.
// ---------------------------------------------------------------------------

typedef __attribute__((ext_vector_type(2))) float v2f;
typedef __attribute__((ext_vector_type(4))) float f4;
typedef __attribute__((ext_vector_type(8))) float v8f;

#define N_NODES 2048
#define FDIM 64
#define ODIM 64

// ---- Kernel 1: P[2048][64] = feat[2048][64] @ weight[0:64][64] -------------
// One wave32 per 16x16 tile of P, K-loop of 16 x V_WMMA_F32_16X16X4_F32.
// A (16x4 f32) layout: lanes 0-15 -> M=lane, v0=K0,v1=K1; lanes 16-31 -> K2,K3.
// B (4x16 f32) layout mirrors A with N=lane&15.
// D (16x16 f32): VGPR r -> M = r + 8*(lane>=16), N = lane&15.
__global__ void edgegcn_p_wmma(const float* __restrict__ feat,
                               const float* __restrict__ weight,
                               float* __restrict__ P) {
  const int lane = threadIdx.x & 31;
  const int wave = threadIdx.x >> 5;
  const int tile = blockIdx.x * 8 + wave;        // 512 tiles: 128 row x 4 col
  const int row0 = (tile >> 2) * 16;
  const int col0 = (tile & 3) * 16;
  const int half = lane >> 4;                    // 0 or 1
  const int l16  = lane & 15;

  v8f acc = {};
#pragma unroll 4
  for (int k0 = 0; k0 < FDIM; k0 += 4) {
    const int ka = k0 + 2 * half;
    v2f a, b;
    a.x = feat[(row0 + l16) * FDIM + ka];
    a.y = feat[(row0 + l16) * FDIM + ka + 1];
    b.x = weight[ka * ODIM + col0 + l16];
    b.y = weight[(ka + 1) * ODIM + col0 + l16];
    acc = __builtin_amdgcn_wmma_f32_16x16x4_f32(
        /*neg_a=*/false, a, /*neg_b=*/false, b,
        /*c_mod=*/(short)0, acc, /*reuse_a=*/false, /*reuse_b=*/false);
  }
#pragma unroll
  for (int r = 0; r < 8; ++r)
    P[(row0 + r + 8 * half) * ODIM + col0 + l16] = acc[r];
}

// ---- Kernel 2: stream the 1 GiB output -------------------------------------
// Block = one source node i. 256 threads = 16 j-slots x 16 col-groups(4 cols).
// Per (j, colgroup): 1 cached b128 load of P[j] (L2-resident), 1 adj scalar,
// 4 FMAs, 1 non-temporal b128 store.
__global__ void edgegcn_stream(const float* __restrict__ adj,
                               const float* __restrict__ weight,
                               const float* __restrict__ bias,
                               const float* __restrict__ P,
                               float* __restrict__ out) {
  const int i  = blockIdx.x;
  const int t  = threadIdx.x;
  const int cg = t & 15;        // which group of 4 output columns
  const int j0 = t >> 4;        // starting j slot, stride 16

  const f4 Pi = ((const f4*)(P + i * ODIM))[cg];
  const f4 Wl = ((const f4*)(weight + FDIM * ODIM))[cg];  // last weight row
  const f4 Bb = ((const f4*)bias)[cg];

  f4 base;
  base.x = Bb.x - Pi.x;
  base.y = Bb.y - Pi.y;
  base.z = Bb.z - Pi.z;
  base.w = Bb.w - Pi.w;

  const float* adj_row = adj + (size_t)i * N_NODES;
  float* out_row = out + (size_t)i * N_NODES * ODIM;

  for (int j = j0; j < N_NODES; j += 16) {
    const float a = adj_row[j];
    const f4 Pj = ((const f4*)(P + j * ODIM))[cg];
    f4 o;
    o.x = __builtin_fmaf(a, Wl.x, Pj.x + base.x);
    o.y = __builtin_fmaf(a, Wl.y, Pj.y + base.y);
    o.z = __builtin_fmaf(a, Wl.z, Pj.z + base.z);
    o.w = __builtin_fmaf(a, Wl.w, Pj.w + base.w);
    __builtin_nontemporal_store(o, &((f4*)(out_row + (size_t)j * ODIM))[cg]);
  }
}

extern "C" void kernel_launch(void* const* d_in, const int* in_sizes, int n_in,
                              void* d_out, int out_size, void* d_ws, size_t ws_size,
                              hipStream_t stream) {
  const float* feat   = (const float*)d_in[0];   // [2048, 64]
  const float* adj    = (const float*)d_in[1];   // [2048, 2048]
  const float* weight = (const float*)d_in[2];   // [65, 64]
  const float* bias   = (const float*)d_in[3];   // [64]
  float* out = (float*)d_out;                    // [2048*2048, 64]
  float* P   = (float*)d_ws;                     // [2048, 64] scratch (512 KB)

  // 512 wave-tiles, 8 waves (256 threads) per block -> 64 blocks.
  edgegcn_p_wmma<<<64, 256, 0, stream>>>(feat, weight, P);
  // One block per source node i.
  edgegcn_stream<<<N_NODES, 256, 0, stream>>>(adj, weight, bias, P, out);
}